// LiquidS4Cell_83537113907226
// MI455X (gfx1250) — compile-verified
//
#include <hip/hip_runtime.h>
#include <hip/hip_bf16.h>

// S4D diagonal SSM -> closed-form depthwise causal convolution, executed on
// the CDNA5 matrix pipe (V_WMMA_F32_16X16X4_F32, wave32).
//
//   a[n]^k           = exp(k * w[n]),  w = -dt*exp(Lambda)      (prep kernel)
//   K[k,d]           = Re( sum_n C[n,d]*Bd[n]*a[n]^k )          (WMMA GEMM)
//   y[b,t,d]         = sum_{k=0..t} K[k,d] * u[b,t-k,d]         (WMMA conv)
//
// BATCH=16 == WMMA M dim; time tiled in 16-step chunks; Toeplitz blocks of K
// form the B matrices. fp32 throughout (reference precision).
//
// Workspace layout (floats): V[512*128] | W[128*256] | Kpad[528*256]
//   (Kpad rows 0..15 are zeros so the diagonal Toeplitz block can read
//    K[t-s] with t-s in [-15,15] without branches)
// Total: 233472 floats = 933888 bytes of d_ws.

#define BATCH   16
#define LEN     512
#define NSTATE  64
#define DMODEL  256
#define NTILE   32           // LEN/16 time chunks
#define UPITCH  516          // LDS row pitch for u tile (516 % 64 == 4 -> conflict-free)

typedef __attribute__((ext_vector_type(2))) float v2f;
typedef __attribute__((ext_vector_type(8))) float v8f;

// ---------------------------------------------------------------------------
// Kernel 1: build V = [re(a^k) | im(a^k)]  (512 x 128),
//                 W = [re(Bd*C) ; -im(Bd*C)]  (128 x 256),
//           zero Kpad rows 0..15.
// ---------------------------------------------------------------------------
__global__ void s4d_prep_kernel(const float* __restrict__ Lre,
                                const float* __restrict__ Lim,
                                const float* __restrict__ Bre,
                                const float* __restrict__ Bim,
                                const float* __restrict__ Pre,
                                const float* __restrict__ Pim,
                                const float* __restrict__ log_dt,
                                float* __restrict__ V,
                                float* __restrict__ W,
                                float* __restrict__ Kpad) {
    const int id = blockIdx.x * blockDim.x + threadIdx.x;
    const float dt = expf(log_dt[0]);

    if (id < LEN * NSTATE) {                      // V: a[n]^k components
        const int k = id >> 6;
        const int n = id & (NSTATE - 1);
        const float e   = expf(Lre[n]);
        const float wre = -dt * e * cosf(Lim[n]); // Re(-dt*exp(Lambda))
        const float wim = -dt * e * sinf(Lim[n]); // Im(-dt*exp(Lambda))
        const float kf  = (float)k;
        const float mag = expf(kf * wre);
        V[k * 128 + n]          = mag * cosf(kf * wim);
        V[k * 128 + NSTATE + n] = mag * sinf(kf * wim);
    } else if (id < LEN * NSTATE + NSTATE * DMODEL) {  // W: Bd*C, imag negated
        const int j = id - LEN * NSTATE;
        const int n = j >> 8;
        const int d = j & (DMODEL - 1);
        const float br = dt * Bre[n];
        const float bi = dt * Bim[n];
        const float pr = Pre[n * DMODEL + d];
        const float pi = Pim[n * DMODEL + d];
        W[n * DMODEL + d]            =  br * pr - bi * pi;   //  Re(Bd*C)
        W[(NSTATE + n) * DMODEL + d] = -(br * pi + bi * pr); // -Im(Bd*C)
    } else {                                      // Kpad rows 0..15 = 0
        const int j = id - (LEN * NSTATE + NSTATE * DMODEL);
        if (j < 16 * DMODEL) Kpad[j] = 0.0f;
    }
}

// ---------------------------------------------------------------------------
// Kernel 2: Kpad[16+k, d] = V[k, :] . W[:, d]   (512x256 GEMM, 128-deep)
// One wave per 16x16 output tile: 32 k-tiles x 16 d-tiles = 512 waves.
// ---------------------------------------------------------------------------
__global__ void s4d_kgemm_kernel(const float* __restrict__ V,
                                 const float* __restrict__ W,
                                 float* __restrict__ Kpad) {
    const int lane = threadIdx.x & 31;
    const int wave = threadIdx.x >> 5;
    const int w    = blockIdx.x * 8 + wave;      // 0..511
    const int kt   = w & (NTILE - 1);            // 0..31
    const int dtl  = w >> 5;                     // 0..15
    const int k0   = kt * 16;
    const int d0   = dtl * 16;

    const int m    = lane & 15;                  // A row (k) and B col (d)
    const int half = lane >> 4;                  // K-slot split across halves

    v8f acc = {};
    #pragma unroll 4
    for (int c0 = 0; c0 < 128; c0 += 4) {
        const int cc = c0 + half * 2;
        v2f a = { V[(k0 + m) * 128 + cc], V[(k0 + m) * 128 + cc + 1] };
        v2f b = { W[cc * DMODEL + d0 + m], W[(cc + 1) * DMODEL + d0 + m] };
        acc = __builtin_amdgcn_wmma_f32_16x16x4_f32(false, a, false, b,
                                                    (short)0, acc, false, false);
    }
    // C layout: VGPR r -> row r + 8*half, col = lane&15
    #pragma unroll
    for (int r = 0; r < 8; ++r) {
        Kpad[(16 + k0 + half * 8 + r) * DMODEL + d0 + m] = acc[r];
    }
}

// ---------------------------------------------------------------------------
// Kernel 3: causal depthwise conv via block-Toeplitz WMMA.
// One block per channel d; u[:,:,d] (16x512) and Kpad[:,d] (528) in LDS.
// 8 waves; wave handles output chunks i = wave + 8r, r=0..3.
// Tile: Y(16b x 16t) += U_j(16b x 16s) x T_{i-j}(16s x 16t), 4 WMMAs per j.
// ---------------------------------------------------------------------------
__global__ void s4d_conv_kernel(const float* __restrict__ u,
                                const float* __restrict__ Kpad,
                                float* __restrict__ out) {
    __shared__ float uS[BATCH * UPITCH];
    __shared__ float KpS[528];

    const int tid  = threadIdx.x;
    const int d    = blockIdx.x;

    // Stage u[b, t, d] -> uS[b*UPITCH + t] (strided gather; all L2 resident)
    for (int it = tid; it < BATCH * LEN; it += 256) {
        const int b = it >> 9;
        const int t = it & (LEN - 1);
        uS[b * UPITCH + t] = u[(b * LEN + t) * DMODEL + d];
    }
    for (int it = tid; it < 528; it += 256) {
        KpS[it] = Kpad[it * DMODEL + d];
    }
    __syncthreads();

    const int lane = tid & 31;
    const int wave = tid >> 5;
    const int m    = lane & 15;   // batch row for A / time col for B
    const int half = lane >> 4;

    for (int r = 0; r < 4; ++r) {
        const int i = wave + 8 * r;              // output time chunk
        v8f acc = {};
        for (int j = 0; j <= i; ++j) {
            const int aBase = m * UPITCH + 16 * j;       // u row m, chunk j
            const int bBase = 16 + 16 * (i - j) + m;     // Kpad[... + t - s]
            #pragma unroll
            for (int s0 = 0; s0 < 16; s0 += 4) {
                const int cc = s0 + half * 2;
                v2f a = { uS[aBase + cc], uS[aBase + cc + 1] };
                v2f b = { KpS[bBase - cc], KpS[bBase - cc - 1] };
                acc = __builtin_amdgcn_wmma_f32_16x16x4_f32(false, a, false, b,
                                                            (short)0, acc,
                                                            false, false);
            }
        }
        // C layout: VGPR rr -> batch = rr + 8*half, time = 16*i + m
        #pragma unroll
        for (int rr = 0; rr < 8; ++rr) {
            const int b = half * 8 + rr;
            out[(b * LEN + 16 * i + m) * DMODEL + d] = acc[rr];
        }
    }
}

// ---------------------------------------------------------------------------
extern "C" void kernel_launch(void* const* d_in, const int* in_sizes, int n_in,
                              void* d_out, int out_size, void* d_ws, size_t ws_size,
                              hipStream_t stream) {
    const float* u      = (const float*)d_in[0];
    const float* Lre    = (const float*)d_in[1];
    const float* Lim    = (const float*)d_in[2];
    const float* Bre    = (const float*)d_in[3];
    const float* Bim    = (const float*)d_in[4];
    const float* Pre    = (const float*)d_in[5];
    const float* Pim    = (const float*)d_in[6];
    const float* log_dt = (const float*)d_in[7];
    float* out = (float*)d_out;

    // Workspace carve-out (needs >= 933888 bytes)
    float* V  = (float*)d_ws;           // 512*128
    float* W  = V + LEN * 128;          // 128*256
    float* Kp = W + 128 * DMODEL;       // 528*256

    // 1) closed-form powers + folded B/C products + pad zeros
    const int prep_elems = LEN * NSTATE + NSTATE * DMODEL + 16 * DMODEL; // 53248
    s4d_prep_kernel<<<prep_elems / 256, 256, 0, stream>>>(
        Lre, Lim, Bre, Bim, Pre, Pim, log_dt, V, W, Kp);

    // 2) K = Re(V_complex @ W_complex) as one 128-deep real WMMA GEMM
    s4d_kgemm_kernel<<<64, 256, 0, stream>>>(V, W, Kp);

    // 3) block-Toeplitz causal conv on the matrix pipe
    s4d_conv_kernel<<<DMODEL, 256, 0, stream>>>(u, Kp, out);
}